// GCN_61108794688062
// MI455X (gfx1250) — compile-verified
//
#include <hip/hip_runtime.h>
#include <hip/hip_bf16.h>

#define NN   50000
#define EE   800000
#define HH   128
#define GG   512
#define CC   10
#define LDS_S 132   // padded LDS row stride (floats) to avoid bank conflicts

typedef __attribute__((ext_vector_type(2))) float v2f;
typedef __attribute__((ext_vector_type(8))) float v8f;

// ---------------- degree / normalization ----------------
__global__ void __launch_bounds__(256) k_set1(float* __restrict__ deg) {
    int i = blockIdx.x * 256 + threadIdx.x;
    if (i < NN) deg[i] = 1.0f;  // self-loop
}

__global__ void __launch_bounds__(256) k_deg(const long long* __restrict__ ei,
                                             float* __restrict__ deg) {
    int e = blockIdx.x * 256 + threadIdx.x;
    if (e < EE) {
        int dst = (int)ei[EE + e];
        unsafeAtomicAdd(&deg[dst], 1.0f);
    }
}

__global__ void __launch_bounds__(256) k_rsqrt(float* __restrict__ dis) {
    int i = blockIdx.x * 256 + threadIdx.x;
    if (i < NN) dis[i] = rsqrtf(dis[i]);
}

// ---------------- WMMA GEMM: Out[N,128] = A[N,128] @ W[128,128] ----------------
// grid.x = N/16 = 3125 blocks, 256 threads = 8 waves; wave w -> output cols [16w,16w+16)
__global__ void __launch_bounds__(256) k_gemm(const float* __restrict__ A,
                                              const float* __restrict__ W,
                                              float* __restrict__ Out) {
    __shared__ float sA[16 * LDS_S];   // 16 rows of A, full K=128
    __shared__ float sW[64 * LDS_S];   // 64-row slab of W (K-half)

    const int tid  = threadIdx.x;
    const int m0   = blockIdx.x * 16;
    const int wave = tid >> 5;
    const int lane = tid & 31;
    const int half = lane >> 4;        // 0 -> K pair {k,k+1}, 1 -> {k+2,k+3}
    const int l    = lane & 15;
    const int n0   = wave * 16;

    // stage A tile (coalesced; shared by all 8 waves)
    for (int i = tid; i < 16 * HH; i += 256) {
        int r = i >> 7, k = i & 127;
        sA[r * LDS_S + k] = A[(size_t)(m0 + r) * HH + k];
    }

    v8f acc = {0.f, 0.f, 0.f, 0.f, 0.f, 0.f, 0.f, 0.f};

    for (int kh = 0; kh < 2; ++kh) {
        __syncthreads();               // protects sA (1st iter) / sW reuse (2nd iter)
        for (int i = tid; i < 64 * HH; i += 256) {
            int k = i >> 7, n = i & 127;
            sW[k * LDS_S + n] = W[(size_t)(kh * 64 + k) * HH + n];
        }
        __syncthreads();

        #pragma unroll
        for (int k0 = 0; k0 < 64; k0 += 4) {
            const int kk = k0 + half * 2;            // local K within slab
            const int ka = kh * 64 + kk;             // global K for A
            v2f a, b;
            a.x = sA[l * LDS_S + ka];
            a.y = sA[l * LDS_S + ka + 1];
            b.x = sW[kk * LDS_S + n0 + l];
            b.y = sW[(kk + 1) * LDS_S + n0 + l];
            acc = __builtin_amdgcn_wmma_f32_16x16x4_f32(
                false, a, false, b, (short)0, acc, false, false);
        }
    }

    // C/D layout: VGPR r -> M = r (lanes 0-15), M = r+8 (lanes 16-31); N = n0+l
    #pragma unroll
    for (int r = 0; r < 8; ++r) {
        Out[(size_t)(m0 + r + half * 8) * HH + n0 + l] = acc[r];
    }
}

// ---------------- self-loop + bias init: agg = h*dis^2 + b ----------------
__global__ void __launch_bounds__(256) k_selfloop(const float* __restrict__ h,
                                                  const float* __restrict__ dis,
                                                  const float* __restrict__ bias,
                                                  float* __restrict__ agg) {
    int t = blockIdx.x * 256 + threadIdx.x;   // N*32 threads, float4 each
    int node = t >> 5, f = (t & 31) * 4;
    float s = dis[node]; s *= s;
    const float4 v  = *(const float4*)(h + (size_t)node * HH + f);
    const float4 bb = *(const float4*)(bias + f);
    float4 o;
    o.x = v.x * s + bb.x; o.y = v.y * s + bb.y;
    o.z = v.z * s + bb.z; o.w = v.w * s + bb.w;
    *(float4*)(agg + (size_t)node * HH + f) = o;
}

// ---------------- edge gather + scatter-add (wave per edge) ----------------
__global__ void __launch_bounds__(256) k_edge(const float* __restrict__ h,
                                              const long long* __restrict__ ei,
                                              const float* __restrict__ dis,
                                              float* __restrict__ agg) {
    int e = blockIdx.x * 8 + (threadIdx.x >> 5);
    if (e >= EE) return;
    int lane = threadIdx.x & 31;
    int src = (int)ei[e];
    int dst = (int)ei[EE + e];
    float coef = dis[src] * dis[dst];
    const float4 v = *(const float4*)(h + (size_t)src * HH + lane * 4);
    float* p = agg + (size_t)dst * HH + lane * 4;
    unsafeAtomicAdd(p + 0, v.x * coef);
    unsafeAtomicAdd(p + 1, v.y * coef);
    unsafeAtomicAdd(p + 2, v.z * coef);
    unsafeAtomicAdd(p + 3, v.w * coef);
}

// ---------------- relu ----------------
__global__ void __launch_bounds__(256) k_relu(float* __restrict__ x) {
    int t = blockIdx.x * 256 + threadIdx.x;   // N*H/4 threads
    float4 v = *(float4*)(x + (size_t)t * 4);
    v.x = fmaxf(v.x, 0.f); v.y = fmaxf(v.y, 0.f);
    v.z = fmaxf(v.z, 0.f); v.w = fmaxf(v.w, 0.f);
    *(float4*)(x + (size_t)t * 4) = v;
}

// ---------------- pooling ----------------
__global__ void __launch_bounds__(256) k_zero(float* __restrict__ pooled,
                                              float* __restrict__ counts) {
    int t = blockIdx.x * 256 + threadIdx.x;
    if (t < GG * HH) pooled[t] = 0.f;
    if (t < GG)      counts[t] = 0.f;
}

__global__ void __launch_bounds__(256) k_pool(const float* __restrict__ h,
                                              const long long* __restrict__ batch,
                                              float* __restrict__ pooled,
                                              float* __restrict__ counts) {
    int t = blockIdx.x * 256 + threadIdx.x;   // N*32 threads, float4 each
    int node = t >> 5, q = t & 31, f = q * 4;
    int g = (int)batch[node];
    const float4 v = *(const float4*)(h + (size_t)node * HH + f);
    float* p = pooled + (size_t)g * HH + f;
    unsafeAtomicAdd(p + 0, v.x);
    unsafeAtomicAdd(p + 1, v.y);
    unsafeAtomicAdd(p + 2, v.z);
    unsafeAtomicAdd(p + 3, v.w);
    if (q == 0) unsafeAtomicAdd(&counts[g], 1.0f);
}

// ---------------- classifier head: out = (pooled/count) @ W_lin + b ----------------
__global__ void __launch_bounds__(256) k_head(const float* __restrict__ pooled,
                                              const float* __restrict__ counts,
                                              const float* __restrict__ Wl,
                                              const float* __restrict__ bl,
                                              float* __restrict__ out) {
    int t = blockIdx.x * 256 + threadIdx.x;
    if (t >= GG * CC) return;
    int g = t / CC, c = t % CC;
    float s = 0.f;
    #pragma unroll 4
    for (int k = 0; k < HH; ++k)
        s += pooled[(size_t)g * HH + k] * Wl[k * CC + c];
    float cnt = counts[g];
    cnt = cnt > 1.f ? cnt : 1.f;
    out[t] = s / cnt + bl[c];
}

extern "C" void kernel_launch(void* const* d_in, const int* in_sizes, int n_in,
                              void* d_out, int out_size, void* d_ws, size_t ws_size,
                              hipStream_t stream) {
    const float*     x     = (const float*)d_in[0];
    const long long* ei    = (const long long*)d_in[1];
    const long long* batch = (const long long*)d_in[2];
    const float* W1 = (const float*)d_in[3];  const float* b1 = (const float*)d_in[4];
    const float* W2 = (const float*)d_in[5];  const float* b2 = (const float*)d_in[6];
    const float* W3 = (const float*)d_in[7];  const float* b3 = (const float*)d_in[8];
    const float* Wl = (const float*)d_in[9];  const float* bl = (const float*)d_in[10];
    float* out = (float*)d_out;

    float* hA     = (float*)d_ws;              // N*H
    float* hB     = hA + (size_t)NN * HH;      // N*H
    float* dis    = hB + (size_t)NN * HH;      // N
    float* pooled = dis + NN;                  // G*H
    float* counts = pooled + (size_t)GG * HH;  // G

    const int nb_n    = (NN + 255) / 256;       // 196
    const int nb_e    = (EE + 255) / 256;       // 3125
    const int nb_gemm = NN / 16;                // 3125
    const int nb_vec  = (NN * 32) / 256;        // 6250
    const int nb_edge = EE / 8;                 // 100000

    // degree -> dis = rsqrt(deg + 1)
    k_set1 <<<nb_n, 256, 0, stream>>>(dis);
    k_deg  <<<nb_e, 256, 0, stream>>>(ei, dis);
    k_rsqrt<<<nb_n, 256, 0, stream>>>(dis);

    // layer 1: h = relu(GCNConv(x, W1, b1))
    k_gemm    <<<nb_gemm, 256, 0, stream>>>(x, W1, hA);
    k_selfloop<<<nb_vec,  256, 0, stream>>>(hA, dis, b1, hB);
    k_edge    <<<nb_edge, 256, 0, stream>>>(hA, ei, dis, hB);
    k_relu    <<<nb_vec,  256, 0, stream>>>(hB);

    // layer 2
    k_gemm    <<<nb_gemm, 256, 0, stream>>>(hB, W2, hA);
    k_selfloop<<<nb_vec,  256, 0, stream>>>(hA, dis, b2, hB);
    k_edge    <<<nb_edge, 256, 0, stream>>>(hA, ei, dis, hB);
    k_relu    <<<nb_vec,  256, 0, stream>>>(hB);

    // layer 3 (no relu)
    k_gemm    <<<nb_gemm, 256, 0, stream>>>(hB, W3, hA);
    k_selfloop<<<nb_vec,  256, 0, stream>>>(hA, dis, b3, hB);
    k_edge    <<<nb_edge, 256, 0, stream>>>(hA, ei, dis, hB);

    // global mean pool + head
    k_zero<<<(GG * HH + 255) / 256, 256, 0, stream>>>(pooled, counts);
    k_pool<<<nb_vec, 256, 0, stream>>>(hB, batch, pooled, counts);
    k_head<<<(GG * CC + 255) / 256, 256, 0, stream>>>(pooled, counts, Wl, bl, out);
}